// GMMLoss_15315853377589
// MI455X (gfx1250) — compile-verified
//
#include <hip/hip_runtime.h>
#include <cstdint>
#include <cstddef>

#define Bn        1048576
#define Kd        32
#define Gn        8
#define SIGMA_F   1.0f
#define TILE_ROWS 64
#define NBLK      1024
#define TILES_PER_BLK (Bn / (NBLK * TILE_ROWS))   // 16

typedef __attribute__((ext_vector_type(2))) float        v2f;
typedef __attribute__((ext_vector_type(8))) float        v8f;
typedef __attribute__((ext_vector_type(4))) unsigned int v4u;
typedef __attribute__((ext_vector_type(4))) int          v4i;
typedef __attribute__((ext_vector_type(8))) int          v8i_t;

// ---------------------------------------------------------------------------
// Kernel 0: zero the workspace accumulators (S[8][32][32], MS[8][32], CNT[8])
// ---------------------------------------------------------------------------
#define WS_FLOATS (Gn * Kd * Kd + Gn * Kd + Gn)   // 8456

__global__ void gmm_zero(float* __restrict__ ws, int n) {
  int i = blockIdx.x * blockDim.x + threadIdx.x;
  if (i < n) ws[i] = 0.0f;
}

// ---------------------------------------------------------------------------
// TDM helpers: issue an async 64x32 fp32 tile load (global -> LDS), and a
// separate TENSORcnt wait so the DMA overlaps WMMA compute (double buffering).
// ---------------------------------------------------------------------------
#if __has_builtin(__builtin_amdgcn_tensor_load_to_lds)

__device__ __forceinline__ void tdm_issue(const float* gsrc, float* smu_dst) {
  unsigned lds_off = (unsigned)(size_t)(smu_dst);              // flat low32 = LDS byte offset
  unsigned long long ga = (unsigned long long)(size_t)gsrc;    // tile start (global bytes)
  v4u g0;
  g0[0] = 1u;                                   // count=1 valid descriptor
  g0[1] = lds_off;                              // lds_addr
  g0[2] = (unsigned)(ga & 0xFFFFFFFFull);       // global_addr[31:0]
  g0[3] = (unsigned)((ga >> 32) & 0x1FFFFFFu)   // global_addr[56:32]
        | (2u << 30);                           // type = 2 ("image")
  v8i_t g1;
  g1[0] = (int)(2u << 16);                      // data_size = 4 bytes
  g1[1] = (int)((unsigned)Kd << 16);            // tensor_dim0 = 32      (bits 79:48 lo16)
  g1[2] = (int)((unsigned)TILE_ROWS << 16);     // tensor_dim1 = 64      (bits 111:80 lo16)
  g1[3] = (int)((unsigned)Kd << 16);            // tensor_dim1 hi16=0 | tile_dim0 = 32
  g1[4] = TILE_ROWS;                            // tile_dim1 = 64 | tile_dim2 = 0
  g1[5] = Kd;                                   // tensor_dim0_stride = 32 elems (lo32)
  g1[6] = 0;
  g1[7] = 0;
  v4i z4 = {0, 0, 0, 0};
#if __has_include(<hip/amd_detail/amd_gfx1250_TDM.h>)
  v8i_t z8 = {0, 0, 0, 0, 0, 0, 0, 0};
  __builtin_amdgcn_tensor_load_to_lds(g0, g1, z4, z4, z8, 0);   // 6-arg toolchain
#else
  __builtin_amdgcn_tensor_load_to_lds(g0, g1, z4, z4, 0);       // 5-arg ROCm 7.2
#endif
}

__device__ __forceinline__ void tdm_wait() {
#if __has_builtin(__builtin_amdgcn_s_wait_tensorcnt)
  __builtin_amdgcn_s_wait_tensorcnt(0);
#else
  asm volatile("s_wait_tensorcnt 0" ::: "memory");
#endif
}

#else   // fallback: cooperative synchronous copy by the issuing wave

__device__ __forceinline__ void tdm_issue(const float* gsrc, float* smu_dst) {
  const float4* src4 = (const float4*)gsrc;
  float4* dst4 = (float4*)smu_dst;
  for (int idx = (int)(threadIdx.x & 31); idx < TILE_ROWS * Kd / 4; idx += 32)
    dst4[idx] = src4[idx];
}
__device__ __forceinline__ void tdm_wait() {}

#endif

// ---------------------------------------------------------------------------
// Kernel 1: segmented second-moment reduction with f32 WMMA.
// 256 threads = 8 waves; wave w accumulates group w only.
// Double-buffered LDS tiles: TDM load of tile t+1 overlaps WMMA on tile t.
// ---------------------------------------------------------------------------
__global__ __launch_bounds__(256) void gmm_partials(
    const float* __restrict__ mu, const int* __restrict__ lab,
    float* __restrict__ S, float* __restrict__ MS, float* __restrict__ CNT)
{
  __shared__ float smu[2][TILE_ROWS * Kd];   // 2 x 8 KB row tiles
  __shared__ int   slab[2][TILE_ROWS];

  const int tid  = threadIdx.x;
  const int lane = tid & 31;
  const int g    = tid >> 5;              // wave id == group id (8 waves, 8 groups)
  const int c    = lane & 15;
  const int half = lane >> 4;

  v8f acc00 = {};                         // S_g block [0:16)x[0:16)
  v8f acc01 = {};                         // S_g block [0:16)x[16:32)  (transpose fills lower)
  v8f acc11 = {};                         // S_g block [16:32)x[16:32)
  float ms0 = 0.0f, ms1 = 0.0f, cnt = 0.0f;

  const int tile0 = blockIdx.x * TILES_PER_BLK;

  // Prologue: stage tile 0 into buffer 0
  {
    const int row0 = tile0 * TILE_ROWS;
    if (tid < 32) { tdm_issue(mu + (size_t)row0 * Kd, smu[0]); tdm_wait(); }
    if (tid < TILE_ROWS) slab[0][tid] = lab[row0 + tid];
  }
  __syncthreads();

  for (int t = 0; t < TILES_PER_BLK; ++t) {
    const int cur = t & 1;

    // Kick off the DMA + label fetch for the next tile before computing.
    if (t + 1 < TILES_PER_BLK) {
      const int row1 = (tile0 + t + 1) * TILE_ROWS;
      if (tid < 32) tdm_issue(mu + (size_t)row1 * Kd, smu[cur ^ 1]);
      if (tid < TILE_ROWS) slab[cur ^ 1][tid] = lab[row1 + tid];
    }

    const float* __restrict__ bufm = smu[cur];
    const int*   __restrict__ bufl = slab[cur];

    #pragma unroll
    for (int s = 0; s < TILE_ROWS; s += 4) {
      // A 16x4 f32 layout: lanes 0-15 -> (r0,r1) in (x,y); lanes 16-31 -> (r2,r3)
      const int ra = s + 2 * half;
      const int rb = ra + 1;
      v2f t0, t1, a0, a1;
      t0.x = bufm[ra * Kd + c];        t0.y = bufm[rb * Kd + c];
      t1.x = bufm[ra * Kd + 16 + c];   t1.y = bufm[rb * Kd + 16 + c];
      const float fa = (bufl[ra] == g) ? 1.0f : 0.0f;
      const float fb = (bufl[rb] == g) ? 1.0f : 0.0f;
      a0.x = t0.x * fa;  a0.y = t0.y * fb;     // column-masked A tiles
      a1.x = t1.x * fa;  a1.y = t1.y * fb;

      acc00 = __builtin_amdgcn_wmma_f32_16x16x4_f32(false, a0, false, t0, (short)0, acc00, false, false);
      acc01 = __builtin_amdgcn_wmma_f32_16x16x4_f32(false, a0, false, t1, (short)0, acc01, false, false);
      acc11 = __builtin_amdgcn_wmma_f32_16x16x4_f32(false, a1, false, t1, (short)0, acc11, false, false);

      ms0 += a0.x + a0.y;                      // masked column sums (col c)
      ms1 += a1.x + a1.y;                      // masked column sums (col 16+c)
      if (c == 0) cnt += fa + fb;
    }

    // Incoming tile must be fully in LDS before anyone reads it next iter.
    if (tid < 32) tdm_wait();
    __syncthreads();
  }

  // Flush partials: C layout VGPR v -> row (v + 8*half), col = c
  #pragma unroll
  for (int v = 0; v < 8; ++v) {
    const int i = v + 8 * half;
    atomicAdd(&S[(g * Kd + i) * Kd + c],            acc00[v]);
    atomicAdd(&S[(g * Kd + i) * Kd + 16 + c],       acc01[v]);
    atomicAdd(&S[(g * Kd + 16 + c) * Kd + i],       acc01[v]);   // symmetric fill
    atomicAdd(&S[(g * Kd + 16 + i) * Kd + 16 + c],  acc11[v]);
  }
  atomicAdd(&MS[g * Kd + c],      ms0);
  atomicAdd(&MS[g * Kd + 16 + c], ms1);
  if (c == 0) atomicAdd(&CNT[g], cnt);
}

// ---------------------------------------------------------------------------
// Kernel 2: finalize. One block, 256 threads (8 waves, one group per wave).
// Cholesky in LDS -> logdet; per-lane column solves -> inverse (in place);
// then 64 threads do the pairwise KL sum.
// ---------------------------------------------------------------------------
__global__ __launch_bounds__(256) void gmm_finalize(
    const float* __restrict__ S, const float* __restrict__ MS,
    const float* __restrict__ CNT, float* __restrict__ out)
{
  __shared__ float A[Gn][Kd][Kd];     // 32 KB: sigma -> Cholesky L -> inverse
  __shared__ float muS[Gn][Kd];
  __shared__ float cntS[Gn];
  __shared__ float ldetS[Gn];
  __shared__ float red[64];

  const int tid  = threadIdx.x;
  const int lane = tid & 31;
  const int g    = tid >> 5;

  if (tid < Gn) cntS[tid] = fmaxf(CNT[tid], 1.0f);
  __syncthreads();
  muS[g][lane] = MS[g * Kd + lane] / cntS[g];
  __syncthreads();

  // sigma_g = SIGMA*I + S_g/n - mu mu^T  (lane = column)
  const float invc = 1.0f / cntS[g];
  for (int i = 0; i < Kd; ++i) {
    float v = S[(g * Kd + i) * Kd + lane] * invc - muS[g][i] * muS[g][lane];
    if (i == lane) v += SIGMA_F;
    A[g][i][lane] = v;
  }
  __syncthreads();

  // In-wave Cholesky (lane = row). Lockstep wave32: stores precede dependent loads.
  float ldet = 0.0f;
  const int r = lane;
  for (int k = 0; k < Kd; ++k) {
    const float akk = A[g][k][k];
    if (r == 0) ldet += logf(akk);
    const float d = sqrtf(akk);
    const float inv_d = 1.0f / d;
    if (r == k) A[g][k][k] = d;
    if (r > k)  A[g][r][k] *= inv_d;
    const float lrk = (r > k) ? A[g][r][k] : 0.0f;
    for (int j = k + 1; j < Kd; ++j) {
      const float ljk = A[g][j][k];
      if (r >= j) A[g][r][j] -= lrk * ljk;
    }
  }
  if (r == 0) ldetS[g] = ldet;

  // Inverse: lane solves L L^T x = e_lane (column `lane` of Sigma^{-1})
  float y[Kd], x[Kd];
  #pragma unroll
  for (int i = 0; i < Kd; ++i) {
    float s = (i == lane) ? 1.0f : 0.0f;
    #pragma unroll
    for (int m = 0; m < Kd; ++m)
      if (m < i) s -= A[g][i][m] * y[m];
    y[i] = s / A[g][i][i];
  }
  #pragma unroll
  for (int i = Kd - 1; i >= 0; --i) {
    float s = y[i];
    #pragma unroll
    for (int m = 0; m < Kd; ++m)
      if (m > i) s -= A[g][m][i] * x[m];
    x[i] = s / A[g][i][i];
  }
  // overwrite A[g] with the inverse (all reads of L above already retired in lockstep)
  #pragma unroll
  for (int i = 0; i < Kd; ++i) A[g][i][lane] = x[i];
  __syncthreads();

  // Pairwise KL, 64 pairs on 64 threads
  float partial = 0.0f;
  if (tid < Gn * Gn) {
    const int i = tid >> 3, j = tid & 7;
    const bool mask = (i < Gn - 1) && (j >= 1) && (i != j);
    if (mask) {
      const float ic = 1.0f / cntS[i];
      float tr = 0.0f;
      for (int a = 0; a < Kd; ++a)
        for (int b = 0; b < Kd; ++b) {
          float sig = S[(i * Kd + a) * Kd + b] * ic - muS[i][a] * muS[i][b];
          if (a == b) sig += SIGMA_F;
          tr += A[j][a][b] * sig;           // tr(inv_j * sig_i), both symmetric
        }
      float mah = 0.0f;
      for (int a = 0; a < Kd; ++a) {
        const float da = muS[j][a] - muS[i][a];
        float acc = 0.0f;
        for (int b = 0; b < Kd; ++b)
          acc += A[j][a][b] * (muS[j][b] - muS[i][b]);
        mah += da * acc;
      }
      const float kl = 0.5f * (tr + mah - (float)Kd + ldetS[j] - ldetS[i]);
      partial = kl * cntS[i] * cntS[j];
    }
  }
  if (tid < 64) red[tid] = partial;
  __syncthreads();
  if (tid == 0) {
    float s = 0.0f;
    for (int q = 0; q < 64; ++q) s += red[q];
    out[0] = s / ((float)Bn * (float)Bn);
  }
}

// ---------------------------------------------------------------------------
extern "C" void kernel_launch(void* const* d_in, const int* in_sizes, int n_in,
                              void* d_out, int out_size, void* d_ws, size_t ws_size,
                              hipStream_t stream) {
  (void)in_sizes; (void)n_in; (void)out_size; (void)ws_size;
  const float* mu  = (const float*)d_in[0];
  const int*   lab = (const int*)d_in[1];
  float* ws  = (float*)d_ws;
  float* S   = ws;                         // [8][32][32]
  float* MS  = ws + Gn * Kd * Kd;          // [8][32]
  float* CNT = MS + Gn * Kd;               // [8]
  float* outp = (float*)d_out;

  gmm_zero<<<(WS_FLOATS + 255) / 256, 256, 0, stream>>>(ws, WS_FLOATS);
  gmm_partials<<<NBLK, 256, 0, stream>>>(mu, lab, S, MS, CNT);
  gmm_finalize<<<1, 256, 0, stream>>>(S, MS, CNT, outp);
}